// LongRangeTransitionLayer_78941498900525
// MI455X (gfx1250) — compile-verified
//
#include <hip/hip_runtime.h>
#include <math.h>

// Problem constants (from reference)
#define H_   16     // NUM_HEADS
#define T_   8      // TRANS_DIM
#define J_   32     // D / H
#define D_   512    // INPUT_DIM
#define ORDER 16    // EXP_ORDER
#define LDS_STRIDE 137   // 136 useful cols + pad (137 coprime-ish with 64 banks)

typedef __attribute__((ext_vector_type(2))) float v2f;
typedef __attribute__((ext_vector_type(8))) float v8f;

// ---- 8x8 truncated matrix exponential of skew(M), Horner form ----
// out = sum_{k=0}^{15} (n^k/k!) * msn^k  where msn = 0.1*(M - M^T)/n,
// n = max(sqrt(sum(skew^2 + 1e-3)), 1)
__device__ __forceinline__ void orth8(const float* __restrict__ Mm,
                                      float* __restrict__ R) {
  float S[64];
  float ss = 0.f;
#pragma unroll
  for (int a = 0; a < 8; ++a)
#pragma unroll
    for (int t = 0; t < 8; ++t) {
      float v = 0.1f * (Mm[a * 8 + t] - Mm[t * 8 + a]);
      S[a * 8 + t] = v;
      ss += v * v + 1e-3f;
    }
  float n = fmaxf(sqrtf(ss), 1.0f);
  float inv = 1.0f / n;
#pragma unroll
  for (int i = 0; i < 64; ++i) S[i] *= inv;  // msn
  // c15 = n^15 / 15!
  float c = 1.0f;
#pragma unroll
  for (int k = 1; k < ORDER; ++k) c *= n / (float)k;
#pragma unroll
  for (int i = 0; i < 64; ++i) R[i] = 0.f;
#pragma unroll
  for (int d = 0; d < 8; ++d) R[d * 9] = c;
  float coef = c;
#pragma unroll 1
  for (int k = ORDER - 2; k >= 0; --k) {
    coef = coef * (float)(k + 1) * inv;  // -> n^k/k!
    float Tm[64];
#pragma unroll
    for (int a = 0; a < 8; ++a)
#pragma unroll
      for (int cc = 0; cc < 8; ++cc) {
        float acc = 0.f;
#pragma unroll
        for (int t = 0; t < 8; ++t) acc = fmaf(S[a * 8 + t], R[t * 8 + cc], acc);
        Tm[a * 8 + cc] = acc;
      }
#pragma unroll
    for (int i = 0; i < 64; ++i) R[i] = Tm[i];
#pragma unroll
    for (int d = 0; d < 8; ++d) R[d * 9] += coef;
  }
}

// ---- power iteration on (A - lam*I), 8 steps ----
__device__ __forceinline__ void powiter(const float* __restrict__ A, float lam,
                                        float* __restrict__ xv) {
#pragma unroll 1
  for (int it = 0; it < 8; ++it) {
    float y[8];
#pragma unroll
    for (int i = 0; i < 8; ++i) {
      float acc = -lam * xv[i];
#pragma unroll
      for (int j = 0; j < 8; ++j) acc = fmaf(A[i * 8 + j], xv[j], acc);
      y[i] = acc;
    }
    float nn = 0.f;
#pragma unroll
    for (int i = 0; i < 8; ++i) nn += y[i] * y[i];
    float r = 1.0f / (sqrtf(nn) + 1e-3f);
#pragma unroll
    for (int i = 0; i < 8; ++i) xv[i] = y[i] * r;
  }
}

__device__ __forceinline__ float quadform(const float* __restrict__ A, float lam,
                                          const float* __restrict__ xv) {
  float s = 0.f;
#pragma unroll
  for (int i = 0; i < 8; ++i) {
    float acc = -lam * xv[i];
#pragma unroll
    for (int j = 0; j < 8; ++j) acc = fmaf(A[i * 8 + j], xv[j], acc);
    s += xv[i] * acc;
  }
  return s;
}

// One block = 64 threads (2 waves) = 64 head-tokens of one head.
// Phase 1: WMMA projection GEMM (32 tokens x 136 cols per wave) -> LDS.
// Phase 2: per-lane 8x8 matrix pipeline -> global out.
__global__ __launch_bounds__(64, 1) void lrt_kernel(
    const float* __restrict__ x, const float* __restrict__ inw,
    const float* __restrict__ inb, const float* __restrict__ outw,
    const float* __restrict__ outb, const float* __restrict__ eigw,
    const float* __restrict__ eigb, float* __restrict__ out, int tokens) {
  __shared__ float smat[64 * LDS_STRIDE];

  const int h = blockIdx.x % H_;
  const int tb = (blockIdx.x / H_) * 64;
  if (tb + 64 > tokens) return;  // uniform guard
  const int tid = threadIdx.x;
  const int wave = tid >> 5;
  const int lane = tid & 31;
  const int half = lane >> 4;   // K lane-half (ISA A/B f32 layout)
  const int lm = lane & 15;     // M row (A) / N col (B,C,D)
  const int wtb = tb + wave * 32;
  const int lrb = wave * 32;

  // ---- Phase 1: A fragments (tokens x K=32), direct float2 loads ----
  v2f afrag[2][8];
#pragma unroll
  for (int m = 0; m < 2; ++m) {
    const float* xr = x + (size_t)(wtb + 16 * m + lm) * D_ + h * J_;
#pragma unroll
    for (int k = 0; k < 8; ++k)
      afrag[m][k] = *(const v2f*)(xr + 4 * k + 2 * half);
  }

#pragma unroll 1
  for (int nt = 0; nt < 9; ++nt) {
    const int n = nt * 16 + lm;  // output column 0..143 (136..143 dead)
    const float* wbase = nullptr;
    float bias = 0.f;
    if (n < 64) {
      wbase = inw + ((size_t)h * 64 + n) * 32;
      bias = inb[h * 64 + n];
    } else if (n < 128) {
      wbase = outw + ((size_t)h * 64 + (n - 64)) * 32;
      bias = outb[h * 64 + (n - 64)];
    } else if (n < 136) {
      wbase = eigw + ((size_t)h * 8 + (n - 128)) * 32;
      bias = eigb[h * 8 + (n - 128)];
    }
    v2f bfrag[8];
#pragma unroll
    for (int k = 0; k < 8; ++k) {
      if (wbase) {
        bfrag[k] = *(const v2f*)(wbase + 4 * k + 2 * half);
      } else {
        bfrag[k][0] = 0.f;
        bfrag[k][1] = 0.f;
      }
    }
    v8f acc0, acc1;
#pragma unroll
    for (int r = 0; r < 8; ++r) { acc0[r] = bias; acc1[r] = bias; }
#pragma unroll
    for (int k = 0; k < 8; ++k) {
      acc0 = __builtin_amdgcn_wmma_f32_16x16x4_f32(
          false, afrag[0][k], false, bfrag[k], (short)0, acc0, false, false);
      acc1 = __builtin_amdgcn_wmma_f32_16x16x4_f32(
          false, afrag[1][k], false, bfrag[k], (short)0, acc1, false, false);
    }
    const bool isEig = (nt == 8);
    if (!isEig || lm < 8) {
#pragma unroll
      for (int r = 0; r < 8; ++r) {
        float v0 = acc0[r], v1 = acc1[r];
        if (isEig) { v0 = tanhf(v0); v1 = tanhf(v1); }
        smat[(lrb + 0 + r + 8 * half) * LDS_STRIDE + n] = v0;
        smat[(lrb + 16 + r + 8 * half) * LDS_STRIDE + n] = v1;
      }
    }
  }
  __syncthreads();

  // ---- Phase 2: one head-token per lane ----
  float* myr = &smat[tid * LDS_STRIDE];

  // in_mat -> in_orth, parked back into LDS cols 0..63
  {
    float Mi[64];
#pragma unroll
    for (int i = 0; i < 64; ++i) Mi[i] = myr[i];
    float Pin[64];
    orth8(Mi, Pin);
#pragma unroll
    for (int i = 0; i < 64; ++i) myr[i] = Pin[i];
  }

  // out_mat -> out_orth (registers), scale rows by eig
  float Pout[64];
  {
    float Mo[64];
#pragma unroll
    for (int i = 0; i < 64; ++i) Mo[i] = myr[64 + i];
    orth8(Mo, Pout);
  }
  {
    float ev[8];
#pragma unroll
    for (int t = 0; t < 8; ++t) ev[t] = myr[128 + t];
#pragma unroll
    for (int t = 0; t < 8; ++t)
#pragma unroll
      for (int c = 0; c < 8; ++c) Pout[t * 8 + c] *= ev[t];
  }

  // transition = in_orth(LDS) @ (diag(eig) @ out_orth)
  float A_[64];
#pragma unroll
  for (int a = 0; a < 8; ++a) {
    float ia[8];
#pragma unroll
    for (int t = 0; t < 8; ++t) ia[t] = myr[a * 8 + t];
#pragma unroll
    for (int c = 0; c < 8; ++c) {
      float acc = 0.f;
#pragma unroll
      for (int t = 0; t < 8; ++t) acc = fmaf(ia[t], Pout[t * 8 + c], acc);
      A_[a * 8 + c] = acc;
    }
  }

  // spectral-radius clamp (matrix_eigval_limit)
  float xv[8];
#pragma unroll
  for (int i = 0; i < 8; ++i) xv[i] = 1.0f;
  powiter(A_, 1.0f, xv);
  float e1 = quadform(A_, 1.0f, xv);
  float oldev = (e1 < 0.f) ? (1.0f + e1) : 1.0f;
  powiter(A_, oldev, xv);
  float e2 = fmaxf(quadform(A_, oldev, xv) + oldev, 1.0f);  // EIGVAL_MINLIMIT
  float s1 = 0.f;
#pragma unroll
  for (int i = 0; i < 8; ++i)
#pragma unroll
    for (int j = 0; j < 8; ++j)
      s1 += fabsf(A_[i * 8 + j] - ((i == j) ? 1.0f : 0.0f));
  const float sc = (s1 < 1e-3f) ? 1.0f : (1.0f / e2);

  // store: out[((b*S+s)*H + h)*64 + i]
  const int gt = tb + tid;
  float4* op = (float4*)(out + ((size_t)gt * H_ + h) * 64);
#pragma unroll
  for (int q = 0; q < 16; ++q) {
    float4 v;
    v.x = A_[q * 4 + 0] * sc;
    v.y = A_[q * 4 + 1] * sc;
    v.z = A_[q * 4 + 2] * sc;
    v.w = A_[q * 4 + 3] * sc;
    op[q] = v;
  }
}

extern "C" void kernel_launch(void* const* d_in, const int* in_sizes, int n_in,
                              void* d_out, int out_size, void* d_ws,
                              size_t ws_size, hipStream_t stream) {
  (void)n_in; (void)out_size; (void)d_ws; (void)ws_size;
  const float* x = (const float*)d_in[0];
  const float* inw = (const float*)d_in[1];
  const float* inb = (const float*)d_in[2];
  const float* outw = (const float*)d_in[3];
  const float* outb = (const float*)d_in[4];
  const float* eigw = (const float*)d_in[5];
  const float* eigb = (const float*)d_in[6];
  float* out = (float*)d_out;
  const int tokens = in_sizes[0] / D_;  // B*S = 4096
  const int blocks = (tokens / 64) * H_;
  lrt_kernel<<<blocks, 64, 0, stream>>>(x, inw, inb, outw, outb, eigw, eigb,
                                        out, tokens);
}